// SSMLayer_7138235646538
// MI455X (gfx1250) — compile-verified
//
#include <hip/hip_runtime.h>
#include <hip/hip_bf16.h>
#include <math.h>

// ---------------------------------------------------------------------------
// Mamba SSM layer for MI455X (gfx1250, wave32, WMMA).
// All four GEMMs use the native fp32 matrix op V_WMMA_F32_16X16X4_F32 so the
// reference's fp32 numerics are preserved while running on the matrix pipe.
// Each wave register-blocks a 16x(16*NT) C strip; per 16-deep K-chunk all
// 4 A-fragments + 4*NT B-fragments are staged into distinct registers first
// (one big load clause), then a burst of 4*NT WMMAs consumes them.  This
// avoids the WAR hazard on a shared A register that would otherwise serialize
// load->WMMA, letting the next chunk's loads overlap the WMMA burst.
// Waves of a block cover consecutive M-tiles of the same N-strip so the B
// stream hits WGP$ for 7 of 8 waves.
// ---------------------------------------------------------------------------

#define DM      1024          // D_MODEL
#define DI      2048          // D_INNER
#define DS      16            // D_STATE
#define DTR     64            // DT_RANK
#define NBATCH  2
#define SEQ     1024
#define MTOT    (NBATCH*SEQ)  // 2048 rows for every GEMM
#define XDBL_W  96            // DT_RANK + 2*D_STATE

typedef __attribute__((ext_vector_type(2))) float v2f;
typedef __attribute__((ext_vector_type(8))) float v8f;

// ---------------------------------------------------------------------------
// C = A · B^T  (A: MxK row-major lda, B: NxK row-major ldb, C: MxN ldc)
// One wave32 computes a 16 x (16*NT) strip via 16x16x4 f32 WMMA.
// EPI==1 applies +bias then softplus (dt projection epilogue).
//
// Fragment layout (ISA 7.12.2, 32-bit operands):
//   A: lane&15 = M row, lane>>4 selects K pair -> contiguous float2 load
//   B: lane&15 = N col, lane>>4 selects K pair -> contiguous float2 load
//   C: VGPR v -> M = v + 8*(lane>>4), N = lane&15
// ---------------------------------------------------------------------------
template<int EPI, int NT>
__global__ __launch_bounds__(256)
void gemm_nt_wmma(const float* __restrict__ A, const float* __restrict__ B,
                  const float* __restrict__ bias, float* __restrict__ C,
                  int M, int N, int K, int lda, int ldb, int ldc)
{
    const int lane = threadIdx.x & 31;
    const int wave = threadIdx.x >> 5;
    const int nMt  = M >> 4;               // # of 16-row M tiles
    const int nNg  = N / (16 * NT);        // # of N strips
    const int tile = blockIdx.x * 8 + wave;
    if (tile >= nMt * nNg) return;

    // consecutive waves -> consecutive M tiles, same N strip (B reuse in WGP$)
    const int m0 = (tile % nMt) << 4;
    const int n0 = (tile / nMt) * (16 * NT);

    const int l    = lane & 15;
    const int koff = (lane >> 4) << 1;     // 0 or 2

    const float* ap = A + (size_t)(m0 + l) * lda + koff;
    const float* bp[NT];
#pragma unroll
    for (int t = 0; t < NT; ++t)
        bp[t] = B + (size_t)(n0 + 16 * t + l) * ldb + koff;

    v8f acc[NT];
#pragma unroll
    for (int t = 0; t < NT; ++t) acc[t] = (v8f){};

    for (int k = 0; k < K; k += 16) {
        // speculative prefetch ahead on the unique (A) stream; OOB is safe
        __builtin_prefetch(ap + k + 512, 0, 0);

        // ---- stage the whole 16-deep K-chunk into distinct registers ----
        v2f af[4];                          // A fragments (reused NT times)
        v2f bf[NT][4];                      // B fragments
#pragma unroll
        for (int u = 0; u < 4; ++u)
            af[u] = *(const v2f*)(ap + k + 4 * u);
#pragma unroll
        for (int t = 0; t < NT; ++t)
#pragma unroll
            for (int u = 0; u < 4; ++u)
                bf[t][u] = *(const v2f*)(bp[t] + k + 4 * u);

        // ---- WMMA burst: 4*NT matrix ops on the staged fragments ----
#pragma unroll
        for (int u = 0; u < 4; ++u)
#pragma unroll
            for (int t = 0; t < NT; ++t)
                acc[t] = __builtin_amdgcn_wmma_f32_16x16x4_f32(
                    false, af[u], false, bf[t][u], (short)0, acc[t],
                    false, false);
    }

    const int rowb = m0 + ((lane >> 4) << 3);
#pragma unroll
    for (int t = 0; t < NT; ++t) {
        const int col = n0 + 16 * t + l;
#pragma unroll
        for (int v = 0; v < 8; ++v) {
            float val = acc[t][v];
            if (EPI == 1) {                               // dt epilogue
                val += bias[col];
                val = (val > 20.0f) ? val : log1pf(__expf(val));   // softplus
            }
            C[(size_t)(rowb + v) * ldc + col] = val;
        }
    }
}

// ---------------------------------------------------------------------------
// Causal depthwise conv (width 4) + bias + SiLU.  xz holds [xc | z] (ld 4096).
// ---------------------------------------------------------------------------
__global__ __launch_bounds__(256)
void conv_silu_kernel(const float* __restrict__ xz, const float* __restrict__ cw,
                      const float* __restrict__ cb, float* __restrict__ xs)
{
    const int idx = blockIdx.x * blockDim.x + threadIdx.x;
    if (idx >= MTOT * DI) return;
    const int ch = idx % DI;
    const int m  = idx / DI;
    const int t  = m % SEQ;

    float acc = cb[ch];
#pragma unroll
    for (int j = 0; j < 4; ++j) {
        const int tt = t - 3 + j;              // causal: taps at t-3..t
        if (tt >= 0)
            acc += cw[ch * 4 + j] * xz[(size_t)(m - 3 + j) * (2 * DI) + ch];
    }
    xs[(size_t)m * DI + ch] = acc / (1.0f + __expf(-acc));   // silu
}

// ---------------------------------------------------------------------------
// Selective-scan: one thread per (batch, channel), 16-state in registers,
// sequential over T (semantically required).  Reads dt from `dty`, writes the
// gated output y back into the same buffer (read-before-write per element).
// ---------------------------------------------------------------------------
__global__ __launch_bounds__(256)
void ssm_scan_kernel(const float* __restrict__ xs, const float* __restrict__ xdbl,
                     const float* __restrict__ A_log, const float* __restrict__ D_skip,
                     const float* __restrict__ xz, float* __restrict__ dty)
{
    const int idx = blockIdx.x * blockDim.x + threadIdx.x;
    if (idx >= NBATCH * DI) return;
    const int ch = idx % DI;
    const int b  = idx / DI;

    float Arow[DS], h[DS];
#pragma unroll
    for (int s = 0; s < DS; ++s) { Arow[s] = -__expf(A_log[ch * DS + s]); h[s] = 0.0f; }
    const float dsk = D_skip[ch];

    for (int t = 0; t < SEQ; ++t) {
        const size_t m  = (size_t)b * SEQ + t;
        const float  dt = dty[m * DI + ch];
        const float  xv = xs [m * DI + ch];
        const float* xd = xdbl + m * XDBL_W;     // [dt_r(64) | Bv(16) | C(16)]
        float y = 0.0f;
#pragma unroll
        for (int s = 0; s < DS; ++s) {
            const float a = __expf(dt * Arow[s]);
            h[s] = a * h[s] + dt * xd[DTR + s] * xv;
            y += h[s] * xd[DTR + DS + s];
        }
        y += xv * dsk;
        const float z = xz[m * (2 * DI) + DI + ch];
        y *= z / (1.0f + __expf(-z));            // * silu(z)
        dty[m * DI + ch] = y;
    }
}

// ---------------------------------------------------------------------------
extern "C" void kernel_launch(void* const* d_in, const int* in_sizes, int n_in,
                              void* d_out, int out_size, void* d_ws, size_t ws_size,
                              hipStream_t stream)
{
    const float* x         = (const float*)d_in[0];
    const float* in_proj_w = (const float*)d_in[1];
    const float* conv_w    = (const float*)d_in[2];
    const float* conv_b    = (const float*)d_in[3];
    const float* x_proj_w  = (const float*)d_in[4];
    const float* dt_proj_w = (const float*)d_in[5];
    const float* dt_proj_b = (const float*)d_in[6];
    const float* A_log     = (const float*)d_in[7];
    const float* D_skip    = (const float*)d_in[8];
    const float* out_proj_w= (const float*)d_in[9];
    float* out = (float*)d_out;

    // workspace layout (floats): xz | xs | x_dbl | dt(->y in-place)  (~65 MB)
    float* xz   = (float*)d_ws;                        // 2048 x 4096
    float* xs   = xz   + (size_t)MTOT * (2 * DI);      // 2048 x 2048
    float* xdbl = xs   + (size_t)MTOT * DI;            // 2048 x 96
    float* dty  = xdbl + (size_t)MTOT * XDBL_W;        // 2048 x 2048

    // 1) xz = x @ in_proj_w^T          (2048 x 4096 x 1024), 16x64 per wave
    {
        int tiles = (MTOT / 16) * ((2 * DI) / 64);
        gemm_nt_wmma<0, 4><<<(tiles + 7) / 8, 256, 0, stream>>>(
            x, in_proj_w, nullptr, xz, MTOT, 2 * DI, DM, DM, DM, 2 * DI);
    }
    // 2) depthwise conv + SiLU -> xs
    conv_silu_kernel<<<(MTOT * DI + 255) / 256, 256, 0, stream>>>(xz, conv_w, conv_b, xs);

    // 3) x_dbl = xs @ x_proj_w^T       (2048 x 96 x 2048), 16x32 per wave
    {
        int tiles = (MTOT / 16) * (XDBL_W / 32);
        gemm_nt_wmma<0, 2><<<(tiles + 7) / 8, 256, 0, stream>>>(
            xs, x_proj_w, nullptr, xdbl, MTOT, XDBL_W, DI, DI, DI, XDBL_W);
    }
    // 4) dt = softplus(dt_r @ dt_proj_w^T + b)   (2048 x 2048 x 64, lda=96)
    {
        int tiles = (MTOT / 16) * (DI / 64);
        gemm_nt_wmma<1, 4><<<(tiles + 7) / 8, 256, 0, stream>>>(
            xdbl, dt_proj_w, dt_proj_b, dty, MTOT, DI, DTR, XDBL_W, DTR, DI);
    }
    // 5) selective scan + D-skip + z-gate (writes y over dt buffer)
    ssm_scan_kernel<<<(NBATCH * DI + 255) / 256, 256, 0, stream>>>(
        xs, xdbl, A_log, D_skip, xz, dty);

    // 6) out = y @ out_proj_w^T        (2048 x 1024 x 2048), 16x64 per wave
    {
        int tiles = (MTOT / 16) * (DM / 64);
        gemm_nt_wmma<0, 4><<<(tiles + 7) / 8, 256, 0, stream>>>(
            dty, out_proj_w, nullptr, out, MTOT, DM, DI, DI, DI, DM);
    }
}